// QuantumAttentionHead_22754736734669
// MI455X (gfx1250) — compile-verified
//
#include <hip/hip_runtime.h>
#include <math.h>

typedef __attribute__((ext_vector_type(2))) float v2f;
typedef __attribute__((ext_vector_type(4))) float v4f;
typedef __attribute__((ext_vector_type(8))) float v8f;

#define WMMA_F32_4(A, B, C) \
  __builtin_amdgcn_wmma_f32_16x16x4_f32(false, (A), false, (B), (short)0, (C), false, false)

// ---------------------------------------------------------------------------
// 4-qubit statevector gates, fully unrolled; masks constant-fold at call sites.
// Qubit q <-> bit (3-q) of the flat index (axis 0 slowest, as in the reference).
// ---------------------------------------------------------------------------
__device__ __forceinline__ void qp(float* re, float* im, float lam, int m) {
  float cl = __cosf(lam), sl = __sinf(lam);
#pragma unroll
  for (int i = 0; i < 16; ++i)
    if (i & m) {
      float r = re[i], s = im[i];
      re[i] = r * cl - s * sl;
      im[i] = r * sl + s * cl;
    }
}

__device__ __forceinline__ void qcx(float* re, float* im, int mc, int mt) {
#pragma unroll
  for (int i = 0; i < 16; ++i)
    if ((i & mc) && !(i & mt)) {
      int j = i | mt;
      float r = re[i], s = im[i];
      re[i] = re[j]; im[i] = im[j];
      re[j] = r;     im[j] = s;
    }
}

__device__ __forceinline__ void qry(float* re, float* im, float th, int m) {
  float c = __cosf(th * 0.5f), s = __sinf(th * 0.5f);
#pragma unroll
  for (int i = 0; i < 16; ++i)
    if (!(i & m)) {
      int j = i | m;
      float r0 = re[i], i0 = im[i], r1 = re[j], i1 = im[j];
      re[i] = c * r0 - s * r1;  im[i] = c * i0 - s * i1;
      re[j] = s * r0 + c * r1;  im[j] = s * i0 + c * i1;
    }
}

__device__ __forceinline__ float qnn_ev(const float* qf, const float* w) {
  float re[16], im[16];
#pragma unroll
  for (int i = 0; i < 16; ++i) { re[i] = 0.25f; im[i] = 0.f; }  // H^4 |0000>
  const float PI_F = 3.14159265358979323846f;
  // feature phases P(2 x_q)
  qp(re, im, 2.f * qf[0], 8);
  qp(re, im, 2.f * qf[1], 4);
  qp(re, im, 2.f * qf[2], 2);
  qp(re, im, 2.f * qf[3], 1);
  // circular ZZ entanglers: pairs (3,0),(0,1),(1,2),(2,3)
  qcx(re, im, 1, 8); qp(re, im, 2.f*(PI_F-qf[3])*(PI_F-qf[0]), 8); qcx(re, im, 1, 8);
  qcx(re, im, 8, 4); qp(re, im, 2.f*(PI_F-qf[0])*(PI_F-qf[1]), 4); qcx(re, im, 8, 4);
  qcx(re, im, 4, 2); qp(re, im, 2.f*(PI_F-qf[1])*(PI_F-qf[2]), 2); qcx(re, im, 4, 2);
  qcx(re, im, 2, 1); qp(re, im, 2.f*(PI_F-qf[2])*(PI_F-qf[3]), 1); qcx(re, im, 2, 1);
  // RealAmplitudes reps=2 (circular)
#pragma unroll
  for (int rep = 0; rep < 2; ++rep) {
    qry(re, im, w[4 * rep + 0], 8);
    qry(re, im, w[4 * rep + 1], 4);
    qry(re, im, w[4 * rep + 2], 2);
    qry(re, im, w[4 * rep + 3], 1);
    qcx(re, im, 1, 8);
    qcx(re, im, 8, 4);
    qcx(re, im, 4, 2);
    qcx(re, im, 2, 1);
  }
  qry(re, im, w[8], 8);
  qry(re, im, w[9], 4);
  qry(re, im, w[10], 2);
  qry(re, im, w[11], 1);
  float ev = 0.f;
#pragma unroll
  for (int i = 0; i < 16; ++i) {
    float p = re[i] * re[i] + im[i] * im[i];
    ev += (__builtin_popcount(i) & 1) ? -p : p;
  }
  return ev;
}

// ---------------------------------------------------------------------------
// Pass 1: per-batch QNN expectation, block-reduced partial sums (deterministic)
// ---------------------------------------------------------------------------
__global__ __launch_bounds__(256) void qah_qnn(const float* __restrict__ x,
                                               const float* __restrict__ Wq,
                                               const float* __restrict__ w,
                                               float* __restrict__ partial) {
  __shared__ float wqm[16];
  __shared__ float red[256];
  if (threadIdx.x < 16) {
    float s = 0.f;
#pragma unroll
    for (int h = 0; h < 16; ++h) s += Wq[threadIdx.x * 16 + h];
    wqm[threadIdx.x] = s * 0.0625f;  // mean over H of Wq row c
  }
  __syncthreads();
  long long b = (long long)blockIdx.x * 256 + threadIdx.x;
  const v4f* xb = (const v4f*)(x + b * 128);  // rows t=0..3, b128 loads
  float qf[4];
#pragma unroll
  for (int t = 0; t < 4; ++t) {
    v4f r0 = xb[t * 4 + 0], r1 = xb[t * 4 + 1], r2 = xb[t * 4 + 2], r3 = xb[t * 4 + 3];
    float s = 0.f;
#pragma unroll
    for (int c = 0; c < 4; ++c) s += r0[c] * wqm[c] + r1[c] * wqm[4 + c] +
                                     r2[c] * wqm[8 + c] + r3[c] * wqm[12 + c];
    qf[t] = tanhf(s);
  }
  red[threadIdx.x] = qnn_ev(qf, w);
  __syncthreads();
#pragma unroll
  for (int st = 128; st > 0; st >>= 1) {
    if (threadIdx.x < st) red[threadIdx.x] += red[threadIdx.x + st];
    __syncthreads();
  }
  if (threadIdx.x == 0) partial[blockIdx.x] = red[0];
}

// ---------------------------------------------------------------------------
// Pass 2: finalize enh = sigmoid(mean_b ev) * quantum_scale
// ---------------------------------------------------------------------------
__global__ __launch_bounds__(256) void qah_reduce(const float* __restrict__ partial,
                                                  int nblocks, float invB,
                                                  const float* __restrict__ qscale,
                                                  float* __restrict__ enh) {
  __shared__ float red[256];
  red[threadIdx.x] = (threadIdx.x < nblocks) ? partial[threadIdx.x] : 0.f;
  __syncthreads();
#pragma unroll
  for (int st = 128; st > 0; st >>= 1) {
    if (threadIdx.x < st) red[threadIdx.x] += red[threadIdx.x + st];
    __syncthreads();
  }
  if (threadIdx.x == 0) {
    float mean = red[0] * invB;
    enh[0] = (1.f / (1.f + __expf(-mean))) * qscale[0];
  }
}

// ---------------------------------------------------------------------------
// Pass 3: fused attention. One wave32 owns TILES consecutive 16x16 tiles
// (2 batches each); W fragments + enh hoisted out of the tile loop.
// All GEMMs via v_wmma_f32_16x16x4_f32 (K=16 -> 4 chained WMMAs each).
// ---------------------------------------------------------------------------
#define ATTN_TILES 8

__global__ __launch_bounds__(256) void qah_attn(const float* __restrict__ x,
                                                const float* __restrict__ Wk,
                                                const float* __restrict__ Wq,
                                                const float* __restrict__ Wv,
                                                const float* __restrict__ enh_p,
                                                float* __restrict__ out) {
  __shared__ float sq[8][16][17];
  __shared__ float sk[8][16][17];
  __shared__ float sv[8][16][17];
  __shared__ float sw[8][16][17];  // softmaxed weights (separate from sq: fewer hazards)

  const int lane = threadIdx.x & 31;
  const int wave = threadIdx.x >> 5;
  const int lo = lane & 15;   // row (A) or column (B/C/D) index
  const int hi = lane >> 4;   // half-wave selects K-pair / row+8
  const long long gw = (long long)blockIdx.x * 8 + wave;
  const float* xw = x + gw * (ATTN_TILES * 256);  // this wave's contiguous x stream
  float* ow = out + gw * (ATTN_TILES * 256);

  // Hoisted: B-fragments of Wk/Wq/Wv (L2-resident) and the enhancement scalar.
  v2f bk[4], bq[4], bv[4];
#pragma unroll
  for (int kc = 0; kc < 4; ++kc) {
    const int r = kc * 4 + hi * 2;
    bk[kc].x = Wk[r * 16 + lo]; bk[kc].y = Wk[(r + 1) * 16 + lo];
    bq[kc].x = Wq[r * 16 + lo]; bq[kc].y = Wq[(r + 1) * 16 + lo];
    bv[kc].x = Wv[r * 16 + lo]; bv[kc].y = Wv[(r + 1) * 16 + lo];
  }
  const float enh = enh_p[0];
  const bool same = ((lo >> 3) == hi);  // column in this row's batch block?

#pragma unroll 1
  for (int t = 0; t < ATTN_TILES; ++t) {
    const float* xt = xw + t * 256;

    // Prefetch next tile while we compute this one (global_prefetch_b8).
    if (t + 1 < ATTN_TILES)
      __builtin_prefetch(xw + (t + 1) * 256 + lane * 8, 0, 1);

    // A-fragments of X: chunk kc holds K = 4kc..4kc+3 (lane pair select via hi)
    v2f ax[4];
#pragma unroll
    for (int kc = 0; kc < 4; ++kc)
      ax[kc] = *(const v2f*)(xt + lo * 16 + kc * 4 + hi * 2);

    // Projections K,Q,V = X @ W
    v8f Kd = {}, Qd = {}, Vd = {};
#pragma unroll
    for (int kc = 0; kc < 4; ++kc) {
      Kd = WMMA_F32_4(ax[kc], bk[kc], Kd);
      Qd = WMMA_F32_4(ax[kc], bq[kc], Qd);
      Vd = WMMA_F32_4(ax[kc], bv[kc], Vd);
    }

    // D-layout -> LDS tiles for re-layout (transpose) into A/B fragments
#pragma unroll
    for (int j = 0; j < 8; ++j) {
      const int row = j + 8 * hi;
      sq[wave][row][lo] = Qd[j];
      sk[wave][row][lo] = Kd[j];
      sv[wave][row][lo] = Vd[j];
    }
    __syncthreads();

    // scores = Q @ K^T on stacked 16x16 (cross-batch blocks masked later)
    v8f S = {};
#pragma unroll
    for (int kc = 0; kc < 4; ++kc) {
      const int c = kc * 4 + hi * 2;
      v2f a; a.x = sq[wave][lo][c]; a.y = sq[wave][lo][c + 1];  // A = Q[m][k]
      v2f b; b.x = sk[wave][lo][c]; b.y = sk[wave][lo][c + 1];  // B = K^T[k][n] = K[n][k]
      S = WMMA_F32_4(a, b, S);
    }

    // causal softmax per row + post-softmax enh on the whole same-batch block
    v8f Wei;
#pragma unroll
    for (int j = 0; j < 8; ++j) {  // row t = j (for both halves)
      const bool valid = same && ((lo & 7) <= j);
      float sc = S[j] * 0.25f;     // * C^-0.5, C=16
      float m = valid ? sc : -__builtin_inff();
      m = fmaxf(m, __shfl_xor(m, 1, 16));
      m = fmaxf(m, __shfl_xor(m, 2, 16));
      m = fmaxf(m, __shfl_xor(m, 4, 16));
      m = fmaxf(m, __shfl_xor(m, 8, 16));
      float e = valid ? __expf(sc - m) : 0.f;
      float sum = e;
      sum += __shfl_xor(sum, 1, 16);
      sum += __shfl_xor(sum, 2, 16);
      sum += __shfl_xor(sum, 4, 16);
      sum += __shfl_xor(sum, 8, 16);
      Wei[j] = e / sum + (same ? enh : 0.f);  // reference adds enh to ALL T*T entries
    }

    // wei -> LDS, then out = wei @ V
#pragma unroll
    for (int j = 0; j < 8; ++j) sw[wave][j + 8 * hi][lo] = Wei[j];
    __syncthreads();

    v8f O = {};
#pragma unroll
    for (int kc = 0; kc < 4; ++kc) {
      const int c = kc * 4 + hi * 2;
      v2f a; a.x = sw[wave][lo][c]; a.y = sw[wave][lo][c + 1];  // A = wei[m][s]
      v2f b; b.x = sv[wave][c][lo]; b.y = sv[wave][c + 1][lo];  // B = V[s][h]
      O = WMMA_F32_4(a, b, O);
    }

    float* ot = ow + t * 256;
#pragma unroll
    for (int j = 0; j < 8; ++j) ot[(j + 8 * hi) * 16 + lo] = O[j];

    // Separate this tile's sv/sw reads from next tile's LDS stores.
    __syncthreads();
  }
}

// ---------------------------------------------------------------------------
extern "C" void kernel_launch(void* const* d_in, const int* in_sizes, int n_in,
                              void* d_out, int out_size, void* d_ws, size_t ws_size,
                              hipStream_t stream) {
  const float* x  = (const float*)d_in[0];
  const float* Wk = (const float*)d_in[1];
  const float* Wq = (const float*)d_in[2];
  const float* Wv = (const float*)d_in[3];
  const float* w  = (const float*)d_in[4];
  const float* qs = (const float*)d_in[5];
  float* out = (float*)d_out;
  float* wsf = (float*)d_ws;
  float* enh = wsf;            // wsf[0]
  float* partial = wsf + 16;   // up to 256 partials

  const long long B = (long long)in_sizes[0] / 128;  // T*C = 8*16
  const int qblocks = (int)(B / 256);
  qah_qnn<<<qblocks, 256, 0, stream>>>(x, Wq, w, partial);
  qah_reduce<<<1, 256, 0, stream>>>(partial, qblocks, 1.0f / (float)B, qs, enh);
  const int ablocks = (int)(B / (2 * 8 * ATTN_TILES));  // 8 waves, TILES tiles, 2 batches
  qah_attn<<<ablocks, 256, 0, stream>>>(x, Wk, Wq, Wv, enh, out);
}